// ForwardDeformer_61632780698146
// MI455X (gfx1250) — compile-verified
//
#include <hip/hip_runtime.h>
#include <math.h>

#define Bq 1
#define Nq 100000
#define Jq 24
#define Iq 11
#define Dq 16
#define Hq 64
#define Wq 64
#define Mq (Dq*Hq*Wq)   // 65536
#define ITERS 20
#define CVG 0.0002f
#define DVG 1.0f
#define DEDUP 0.01f

typedef float v2f __attribute__((ext_vector_type(2)));
typedef float v8f __attribute__((ext_vector_type(8)));

// ---------------------------------------------------------------------------
// Kernel 1: per-voxel softmax over J=24 logits (scaled by 20).
// lbs_logits layout: [J][M] (j*65536 + m), coalesced per-j loads.
// ---------------------------------------------------------------------------
__global__ void softmax_w_kernel(const float* __restrict__ logits,
                                 float* __restrict__ w) {
    int m = blockIdx.x * blockDim.x + threadIdx.x;
    if (m >= Mq) return;
    float v[Jq];
    float mx = -1e30f;
#pragma unroll
    for (int j = 0; j < Jq; ++j) {
        float t = logits[j * Mq + m] * 20.0f;
        v[j] = t;
        mx = fmaxf(mx, t);
    }
    float s = 0.0f;
#pragma unroll
    for (int j = 0; j < Jq; ++j) {
        v[j] = expf(v[j] - mx);
        s += v[j];
    }
    float inv = 1.0f / s;
#pragma unroll
    for (int j = 0; j < Jq; ++j) {
        w[m * Jq + j] = v[j] * inv;
    }
}

// ---------------------------------------------------------------------------
// Kernel 2: T[m][16] = sum_j w[m][j] * tfs_flat[j][16]  via V_WMMA_F32_16X16X4_F32
// One wave per 16-voxel tile. A = 16x4 f32 tile of w (lanes 0-15: M rows,
// VGPR0 = K0|K2, VGPR1 = K1|K3). B = 4x16 tile of tfs (rows striped across
// lanes). C/D = 16x16 f32 accumulator (8 VGPRs). K=24 done in 6 WMMA steps.
// ---------------------------------------------------------------------------
__global__ void wmma_T_kernel(const float* __restrict__ w,
                              const float* __restrict__ tfs,
                              float* __restrict__ T) {
    int lane = threadIdx.x & 31;
    int wave = threadIdx.x >> 5;
    int tile = blockIdx.x * (blockDim.x >> 5) + wave;
    int m0 = tile * 16;
    if (m0 >= Mq) return;
    int mr = lane & 15;           // row (A: M index, B: N index, D: N index)
    int kh = (lane >> 4) * 2;     // lanes 16-31 hold K+2 / K+3

    v8f c = {};
#pragma unroll
    for (int kc = 0; kc < Jq; kc += 4) {
        v2f a, b;
        a[0] = w[(m0 + mr) * Jq + kc + kh + 0];
        a[1] = w[(m0 + mr) * Jq + kc + kh + 1];
        b[0] = tfs[(kc + kh + 0) * 16 + mr];
        b[1] = tfs[(kc + kh + 1) * 16 + mr];
        c = __builtin_amdgcn_wmma_f32_16x16x4_f32(
                /*neg_a=*/false, a, /*neg_b=*/false, b,
                /*c_mod=*/(short)0, c, /*reuse_a=*/false, /*reuse_b=*/false);
    }
    // D layout: VGPR r -> lanes 0-15: M=r, lanes 16-31: M=r+8; N = lane&15
    int mofs = (lane >> 4) * 8;
#pragma unroll
    for (int r = 0; r < 8; ++r) {
        T[(m0 + r + mofs) * 16 + mr] = c[r];
    }
}

// ---------------------------------------------------------------------------
// 3x3 inverse helper (adjugate / det), producing affine inverse rows [Rinv|-Rinv t]
// ---------------------------------------------------------------------------
__device__ __forceinline__ void affine_inverse_3x4(const float R[3][3], const float t[3],
                                                   float inv[3][4]) {
    float c00 = R[1][1]*R[2][2] - R[1][2]*R[2][1];
    float c01 = R[0][2]*R[2][1] - R[0][1]*R[2][2];
    float c02 = R[0][1]*R[1][2] - R[0][2]*R[1][1];
    float c10 = R[1][2]*R[2][0] - R[1][0]*R[2][2];
    float c11 = R[0][0]*R[2][2] - R[0][2]*R[2][0];
    float c12 = R[0][2]*R[1][0] - R[0][0]*R[1][2];
    float c20 = R[1][0]*R[2][1] - R[1][1]*R[2][0];
    float c21 = R[0][1]*R[2][0] - R[0][0]*R[2][1];
    float c22 = R[0][0]*R[1][1] - R[0][1]*R[1][0];
    float det = R[0][0]*c00 + R[0][1]*c10 + R[0][2]*c20;
    float id = 1.0f / det;
    inv[0][0] = c00*id; inv[0][1] = c01*id; inv[0][2] = c02*id;
    inv[1][0] = c10*id; inv[1][1] = c11*id; inv[1][2] = c12*id;
    inv[2][0] = c20*id; inv[2][1] = c21*id; inv[2][2] = c22*id;
#pragma unroll
    for (int r = 0; r < 3; ++r)
        inv[r][3] = -(inv[r][0]*t[0] + inv[r][1]*t[1] + inv[r][2]*t[2]);
}

// ---------------------------------------------------------------------------
// Kernel 3: per-voxel: vox_d = R*g + t ; Jinv = affine inverse rows.
// Pack 16 floats per voxel: [0..2]=pos, [3]=0, [4..7]=Jrow0(J00,J01,J02,0),
// [8..11]=Jrow1, [12..15]=Jrow2 -> 4 x float4 per corner fetch.
// ---------------------------------------------------------------------------
__global__ void invert_pack_kernel(const float* __restrict__ T,
                                   const float* __restrict__ offset,
                                   const float* __restrict__ scale,
                                   float* __restrict__ pack) {
    int m = blockIdx.x * blockDim.x + threadIdx.x;
    if (m >= Mq) return;
    float R[3][3], t[3];
#pragma unroll
    for (int r = 0; r < 3; ++r) {
#pragma unroll
        for (int c = 0; c < 3; ++c) R[r][c] = T[m * 16 + r * 4 + c];
        t[r] = T[m * 16 + r * 4 + 3];
    }
    float s = scale[0];
    float ox = offset[0], oy = offset[1], oz = offset[2];
    int x = m & 63, y = (m >> 6) & 63, z = m >> 12;
    float gx = -1.0f + 2.0f * (float)x / (float)(Wq - 1);
    float gy = -1.0f + 2.0f * (float)y / (float)(Hq - 1);
    float gz = -1.0f + 2.0f * (float)z / (float)(Dq - 1);
    float px = gx * s + ox;
    float py = gy * s + oy;
    float pz = (gz * 0.25f) * s + oz;   // ratio = H/D = 4
    float vx = R[0][0]*px + R[0][1]*py + R[0][2]*pz + t[0];
    float vy = R[1][0]*px + R[1][1]*py + R[1][2]*pz + t[1];
    float vz = R[2][0]*px + R[2][1]*py + R[2][2]*pz + t[2];
    float inv[3][4];
    affine_inverse_3x4(R, t, inv);
    float* o = pack + m * 16;
    o[0] = vx; o[1] = vy; o[2] = vz; o[3] = 0.0f;
#pragma unroll
    for (int r = 0; r < 3; ++r) {
        o[4 + 4*r + 0] = inv[r][0];
        o[4 + 4*r + 1] = inv[r][1];
        o[4 + 4*r + 2] = inv[r][2];
        o[4 + 4*r + 3] = 0.0f;
    }
}

// ---------------------------------------------------------------------------
// Kernel: inverse of the 11 init-bone transforms -> [11][12] (3x4 rows)
// ---------------------------------------------------------------------------
__global__ void tbinv_kernel(const float* __restrict__ tfs,
                             const int* __restrict__ init_bones,
                             float* __restrict__ out) {
    int i = threadIdx.x;
    if (i >= Iq) return;
    int j = init_bones[i];
    float R[3][3], t[3];
#pragma unroll
    for (int r = 0; r < 3; ++r) {
#pragma unroll
        for (int c = 0; c < 3; ++c) R[r][c] = tfs[j * 16 + r * 4 + c];
        t[r] = tfs[j * 16 + r * 4 + 3];
    }
    float inv[3][4];
    affine_inverse_3x4(R, t, inv);
#pragma unroll
    for (int r = 0; r < 3; ++r)
#pragma unroll
        for (int c = 0; c < 4; ++c) out[i * 12 + r * 4 + c] = inv[r][c];
}

// ---------------------------------------------------------------------------
// Trilinear setup shared by Newton + final validity
// ---------------------------------------------------------------------------
struct Tri {
    int base, sx, sy, sz;
    float w[8];
};

__device__ __forceinline__ Tri tri_setup(float xn0, float xn1, float xn2) {
    float cx = fminf(fmaxf((xn0 + 1.0f) * 0.5f * (float)(Wq - 1), 0.0f), (float)(Wq - 1));
    float cy = fminf(fmaxf((xn1 + 1.0f) * 0.5f * (float)(Hq - 1), 0.0f), (float)(Hq - 1));
    float cz = fminf(fmaxf((xn2 + 1.0f) * 0.5f * (float)(Dq - 1), 0.0f), (float)(Dq - 1));
    int x0 = (int)floorf(cx);
    int y0 = (int)floorf(cy);
    int z0 = (int)floorf(cz);
    int x1 = min(x0 + 1, Wq - 1);
    int y1 = min(y0 + 1, Hq - 1);
    int z1 = min(z0 + 1, Dq - 1);
    float fx = cx - (float)x0;
    float fy = cy - (float)y0;
    float fz = cz - (float)z0;
    Tri t;
    t.base = z0 * (Hq * Wq) + y0 * Wq + x0;
    t.sx = x1 - x0;
    t.sy = (y1 - y0) * Wq;
    t.sz = (z1 - z0) * (Hq * Wq);
    float gx0 = 1.0f - fx, gy0 = 1.0f - fy, gz0 = 1.0f - fz;
    t.w[0] = gz0 * gy0 * gx0; t.w[1] = gz0 * gy0 * fx;
    t.w[2] = gz0 * fy  * gx0; t.w[3] = gz0 * fy  * fx;
    t.w[4] = fz  * gy0 * gx0; t.w[5] = fz  * gy0 * fx;
    t.w[6] = fz  * fy  * gx0; t.w[7] = fz  * fy  * fx;
    return t;
}

// ---------------------------------------------------------------------------
// Kernel 4: Newton iterations. One thread per (n, i). 20 fixed iterations.
// ---------------------------------------------------------------------------
__global__ void newton_kernel(const float* __restrict__ xd,
                              const unsigned char* __restrict__ mask,
                              const float* __restrict__ tbinv,
                              const float* __restrict__ pack,
                              const float* __restrict__ offset,
                              const float* __restrict__ scale,
                              float* __restrict__ out_xc,
                              int* __restrict__ out_valid) {
    int tid = blockIdx.x * blockDim.x + threadIdx.x;
    if (tid >= Nq * Iq) return;
    int n = tid / Iq;
    int i = tid - n * Iq;

    float xdx = xd[n * 3 + 0], xdy = xd[n * 3 + 1], xdz = xd[n * 3 + 2];
    bool base_valid = mask[n] != 0;

    float s = scale[0];
    float ox = offset[0], oy = offset[1], oz = offset[2];
    float skx = 1.0f / s, sky = 1.0f / s, skz = 4.0f / s;  // [1,1,ratio]/scale

    const float* M = tbinv + i * 12;
    float xcx = M[0]*xdx + M[1]*xdy + M[2]*xdz + M[3];
    float xcy = M[4]*xdx + M[5]*xdy + M[6]*xdz + M[7];
    float xcz = M[8]*xdx + M[9]*xdy + M[10]*xdz + M[11];

    const float4* vol4 = (const float4*)pack;

    for (int it = 0; it < ITERS; ++it) {
        Tri tr = tri_setup((xcx - ox) * skx, (xcy - oy) * sky, (xcz - oz) * skz);
        float acc[16];
#pragma unroll
        for (int k = 0; k < 16; ++k) acc[k] = 0.0f;
#pragma unroll
        for (int c = 0; c < 8; ++c) {
            int m = tr.base + ((c & 1) ? tr.sx : 0) + ((c & 2) ? tr.sy : 0) + ((c & 4) ? tr.sz : 0);
            float wgt = tr.w[c];
            const float4* q = vol4 + m * 4;
            float4 q0 = q[0], q1 = q[1], q2 = q[2], q3 = q[3];
            acc[0]  += wgt * q0.x; acc[1]  += wgt * q0.y; acc[2]  += wgt * q0.z;
            acc[4]  += wgt * q1.x; acc[5]  += wgt * q1.y; acc[6]  += wgt * q1.z;
            acc[8]  += wgt * q2.x; acc[9]  += wgt * q2.y; acc[10] += wgt * q2.z;
            acc[12] += wgt * q3.x; acc[13] += wgt * q3.y; acc[14] += wgt * q3.z;
        }
        float ex = acc[0] - xdx, ey = acc[1] - xdy, ez = acc[2] - xdz;
        float en = sqrtf(ex * ex + ey * ey + ez * ez);
        float dlx = -(acc[4]  * ex + acc[5]  * ey + acc[6]  * ez);
        float dly = -(acc[8]  * ex + acc[9]  * ey + acc[10] * ez);
        float dlz = -(acc[12] * ex + acc[13] * ey + acc[14] * ez);
        bool step = (en > CVG) && (en < DVG) && base_valid;
        if (step) { xcx += dlx; xcy += dly; xcz += dlz; }
    }

    // final error -> validity (pos only needed)
    Tri tr = tri_setup((xcx - ox) * skx, (xcy - oy) * sky, (xcz - oz) * skz);
    float px = 0.0f, py = 0.0f, pz = 0.0f;
#pragma unroll
    for (int c = 0; c < 8; ++c) {
        int m = tr.base + ((c & 1) ? tr.sx : 0) + ((c & 2) ? tr.sy : 0) + ((c & 4) ? tr.sz : 0);
        float wgt = tr.w[c];
        float4 q0 = vol4[m * 4];
        px += wgt * q0.x; py += wgt * q0.y; pz += wgt * q0.z;
    }
    float ex = px - xdx, ey = py - xdy, ez = pz - xdz;
    float en = sqrtf(ex * ex + ey * ey + ez * ez);
    bool is_valid = base_valid && (en < CVG);

    out_xc[tid * 3 + 0] = xcx;
    out_xc[tid * 3 + 1] = xcy;
    out_xc[tid * 3 + 2] = xcz;
    out_valid[tid] = is_valid ? 1 : 0;
}

// ---------------------------------------------------------------------------
// Kernel 5: sequential dedup across the 11 inits of each point.
// ---------------------------------------------------------------------------
__global__ void dedup_kernel(const float* __restrict__ xc,
                             const int* __restrict__ valid,
                             float* __restrict__ out_valid) {
    int n = blockIdx.x * blockDim.x + threadIdx.x;
    if (n >= Nq) return;
    float px[Iq], py[Iq], pz[Iq];
    int v[Iq], vn[Iq];
#pragma unroll
    for (int i = 0; i < Iq; ++i) {
        px[i] = xc[(n * Iq + i) * 3 + 0];
        py[i] = xc[(n * Iq + i) * 3 + 1];
        pz[i] = xc[(n * Iq + i) * 3 + 2];
        v[i] = valid[n * Iq + i];
    }
    vn[0] = v[0];
#pragma unroll
    for (int i = 1; i < Iq; ++i) {
        bool dup = false;
#pragma unroll
        for (int k = 0; k < Iq; ++k) {
            if (k >= i) break;
            float dx = px[i] - px[k], dy = py[i] - py[k], dz = pz[i] - pz[k];
            float d2 = dx * dx + dy * dy + dz * dz;
            if (vn[k] && (d2 < DEDUP * DEDUP)) dup = true;
        }
        vn[i] = (v[i] && !dup) ? 1 : 0;
    }
#pragma unroll
    for (int i = 0; i < Iq; ++i) {
        out_valid[n * Iq + i] = vn[i] ? 1.0f : 0.0f;
    }
}

// ---------------------------------------------------------------------------
extern "C" void kernel_launch(void* const* d_in, const int* in_sizes, int n_in,
                              void* d_out, int out_size, void* d_ws, size_t ws_size,
                              hipStream_t stream) {
    const float*         xd        = (const float*)d_in[0];
    const unsigned char* mask      = (const unsigned char*)d_in[1];
    const float*         tfs       = (const float*)d_in[2];
    const float*         logits    = (const float*)d_in[3];
    const float*         offset    = (const float*)d_in[4];
    const float*         scale     = (const float*)d_in[5];
    const int*           init_bones= (const int*)d_in[6];

    // workspace carving (all 256B aligned)
    char* p = (char*)d_ws;
    float* ws_w     = (float*)p; p += (size_t)Mq * Jq * sizeof(float);        // 6.29 MB
    float* ws_T     = (float*)p; p += (size_t)Mq * 16 * sizeof(float);        // 4.19 MB
    float* ws_pack  = (float*)p; p += (size_t)Mq * 16 * sizeof(float);        // 4.19 MB
    float* ws_tbinv = (float*)p; p += 256;                                    // 11*12*4 = 528 -> pad
    p += 256;
    int*   ws_valid = (int*)p;   p += (size_t)Nq * Iq * sizeof(int);          // 4.4 MB

    float* out_xc    = (float*)d_out;                 // (1,N,I,3) flat
    float* out_valid = (float*)d_out + (size_t)Nq * Iq * 3;

    // 1) softmax weights
    softmax_w_kernel<<<(Mq + 255) / 256, 256, 0, stream>>>(logits, ws_w);

    // 2) WMMA GEMM: T = w @ tfs   (8 waves / block -> 8 tiles / block)
    wmma_T_kernel<<<(Mq / 16) / 8, 256, 0, stream>>>(ws_w, tfs, ws_T);

    // 3) per-voxel invert + pack
    invert_pack_kernel<<<(Mq + 255) / 256, 256, 0, stream>>>(ws_T, offset, scale, ws_pack);

    // 3b) init-bone inverses
    tbinv_kernel<<<1, 16, 0, stream>>>(tfs, init_bones, ws_tbinv);

    // 4) Newton iterations
    int ni = Nq * Iq;
    newton_kernel<<<(ni + 255) / 256, 256, 0, stream>>>(xd, mask, ws_tbinv, ws_pack,
                                                        offset, scale, out_xc, ws_valid);

    // 5) dedup -> final validity
    dedup_kernel<<<(Nq + 255) / 256, 256, 0, stream>>>(out_xc, ws_valid, out_valid);
}